// ValueAdvantage_47656957116636
// MI455X (gfx1250) — compile-verified
//
#include <hip/hip_runtime.h>
#include <hip/hip_bf16.h>

typedef __attribute__((ext_vector_type(2))) float v2f;
typedef __attribute__((ext_vector_type(8))) float v8f;

#define N_NODES 100000
#define N_EDGES 3200000
#define BN_EPS  1e-5f

// ---------------- degree / norm ----------------
__global__ void degree_kernel(const int* __restrict__ src, const int* __restrict__ dst,
                              int* __restrict__ deg_out, int* __restrict__ deg_in, int n) {
    int i = blockIdx.x * blockDim.x + threadIdx.x;
    if (i < n) {
        atomicAdd(&deg_out[src[i]], 1);
        atomicAdd(&deg_in[dst[i]], 1);
    }
}

__global__ void dnorm_kernel(const int* __restrict__ deg_out, const int* __restrict__ deg_in,
                             float* __restrict__ dnorm_out, float* __restrict__ dnorm_in, int n) {
    int i = blockIdx.x * blockDim.x + threadIdx.x;
    if (i < n) {
        dnorm_out[i] = rsqrtf(fmaxf((float)deg_out[i], 1.0f));
        dnorm_in[i]  = rsqrtf(fmaxf((float)deg_in[i],  1.0f));
    }
}

// ---------------- SpMM scatter-add: one wave per edge ----------------
// agg[dst] += X[src] * dnorm_out[src]
template <int F>
__global__ void spmm_scatter(const float* __restrict__ X, const int* __restrict__ src,
                             const int* __restrict__ dst, const float* __restrict__ dnorm_out,
                             float* __restrict__ agg, int nEdges) {
    int wid  = (blockIdx.x * blockDim.x + threadIdx.x) >> 5;
    int lane = threadIdx.x & 31;
    if (wid >= nEdges) return;
    int s = src[wid], d = dst[wid];
    float sc = dnorm_out[s];
    const float4* xs = (const float4*)(X + (size_t)s * F);
    float* ad = agg + (size_t)d * F;
#pragma unroll
    for (int c = lane; c < F / 4; c += 32) {
        float4 v = xs[c];
        unsafeAtomicAdd(ad + 4 * c + 0, v.x * sc);
        unsafeAtomicAdd(ad + 4 * c + 1, v.y * sc);
        unsafeAtomicAdd(ad + 4 * c + 2, v.z * sc);
        unsafeAtomicAdd(ad + 4 * c + 3, v.w * sc);
    }
}

// ---------------- in-place row scale (agg *= dnorm_in[row]) ----------------
template <int F>
__global__ void rowscale_kernel(float* __restrict__ X, const float* __restrict__ dnorm, int nRows) {
    int i = blockIdx.x * blockDim.x + threadIdx.x;   // float4 index
    int total = nRows * (F / 4);
    if (i < total) {
        int row = i / (F / 4);
        float s = dnorm[row];
        float4* p = (float4*)X + i;
        float4 v = *p;
        v.x *= s; v.y *= s; v.z *= s; v.w *= s;
        *p = v;
    }
}

// ---------------- WMMA fp32 GEMM: Y[M x NOUT] = X[M x K] @ W[K x NOUT] + bias ----------------
// One wave computes one 16x16 tile using V_WMMA_F32_16X16X4_F32.
template <int K, int NOUT>
__global__ void wmma_gemm_bias(const float* __restrict__ X, const float* __restrict__ W,
                               const float* __restrict__ bias, float* __restrict__ Y, int M) {
    const int NT = NOUT / 16;
    int wid  = (blockIdx.x * blockDim.x + threadIdx.x) >> 5;
    int lane = threadIdx.x & 31;
    int mtile = wid / NT;
    int ntile = wid % NT;
    if (mtile * 16 >= M) return;   // wave-uniform: EXEC stays all-ones for WMMA

    int rsel  = lane & 15;
    int khalf = (lane >> 4) * 2;           // lanes 0-15 -> K 0,1 ; lanes 16-31 -> K 2,3
    int row = mtile * 16 + rsel;           // A-frag row
    int col = ntile * 16 + rsel;           // B-frag / D column

    const float* xrow = X + (size_t)row * K;
    v8f acc = {};
    for (int k = 0; k < K; k += 4) {
        v2f a, b;
        a.x = xrow[k + khalf];
        a.y = xrow[k + khalf + 1];
        b.x = W[(size_t)(k + khalf)     * NOUT + col];
        b.y = W[(size_t)(k + khalf + 1) * NOUT + col];
        acc = __builtin_amdgcn_wmma_f32_16x16x4_f32(false, a, false, b, (short)0, acc,
                                                    false, false);
    }

    float bb = bias[col];
    int rbase = mtile * 16 + ((lane >> 4) << 3);   // lanes 16-31 write rows +8
#pragma unroll
    for (int i = 0; i < 8; ++i) {
        Y[(size_t)(rbase + i) * NOUT + col] = acc[i] + bb;
    }
}

// ---------------- BatchNorm stats: per-column sum & sumsq ----------------
template <int C, int ROWS>
__global__ void bn_stats_kernel(const float* __restrict__ X, float* __restrict__ sums, int M) {
    int col = threadIdx.x;                 // blockDim.x == C
    int r0 = blockIdx.x * ROWS;
    int r1 = r0 + ROWS; if (r1 > M) r1 = M;
    float s = 0.f, q = 0.f;
    for (int r = r0; r < r1; ++r) {
        float v = X[(size_t)r * C + col];
        s += v; q += v * v;
    }
    unsafeAtomicAdd(&sums[col], s);
    unsafeAtomicAdd(&sums[C + col], q);
}

__global__ void bn_coef_kernel(const float* __restrict__ sums, const float* __restrict__ gamma,
                               const float* __restrict__ beta, float* __restrict__ coef,
                               int M, int C) {
    int c = threadIdx.x;
    if (c < C) {
        float invM = 1.0f / (float)M;
        float mean = sums[c] * invM;
        float var  = sums[C + c] * invM - mean * mean;   // biased var, matches reference
        float sc = gamma[c] * rsqrtf(var + BN_EPS);
        coef[c]     = sc;
        coef[C + c] = beta[c] - mean * sc;
    }
}

template <int C>
__global__ void bn_apply_relu(float* __restrict__ X, const float* __restrict__ coef, int M) {
    int i = blockIdx.x * blockDim.x + threadIdx.x;
    if (i < M * C) {
        int c = i & (C - 1);
        float v = X[i] * coef[c] + coef[C + c];
        X[i] = fmaxf(v, 0.0f);
    }
}

// ---------------- dueling head: val + adv - mean(adv), one wave per node ----------------
__global__ void head_kernel(const float* __restrict__ H, const float* __restrict__ adv,
                            const float* __restrict__ Wv, const float* __restrict__ bv,
                            float* __restrict__ Q, int M) {
    int node = (blockIdx.x * blockDim.x + threadIdx.x) >> 5;
    int lane = threadIdx.x & 31;
    if (node >= M) return;
    const float* h = H + (size_t)node * 256;
    float v = 0.f;
#pragma unroll
    for (int k = lane; k < 256; k += 32) v += h[k] * Wv[k];
#pragma unroll
    for (int off = 16; off; off >>= 1) v += __shfl_xor(v, off, 32);
    float val = v + bv[0];

    const float* arow = adv + (size_t)node * 64;
    float a0 = arow[lane], a1 = arow[lane + 32];
    float s = a0 + a1;
#pragma unroll
    for (int off = 16; off; off >>= 1) s += __shfl_xor(s, off, 32);
    float mean = s * (1.0f / 64.0f);

    float* qrow = Q + (size_t)node * 64;
    qrow[lane]      = val + a0 - mean;
    qrow[lane + 32] = val + a1 - mean;
}

// ---------------- launch ----------------
extern "C" void kernel_launch(void* const* d_in, const int* in_sizes, int n_in,
                              void* d_out, int out_size, void* d_ws, size_t ws_size,
                              hipStream_t stream) {
    const float* feat = (const float*)d_in[0];
    const int*   src  = (const int*)  d_in[1];
    const int*   dst  = (const int*)  d_in[2];
    const float* W1 = (const float*)d_in[3];  const float* b1  = (const float*)d_in[4];
    const float* g1 = (const float*)d_in[5];  const float* be1 = (const float*)d_in[6];
    const float* W2 = (const float*)d_in[7];  const float* b2  = (const float*)d_in[8];
    const float* g2 = (const float*)d_in[9];  const float* be2 = (const float*)d_in[10];
    const float* Wa = (const float*)d_in[11]; const float* ba  = (const float*)d_in[12];
    const float* Wv = (const float*)d_in[13]; const float* bv  = (const float*)d_in[14];
    float* Q = (float*)d_out;

    const int N = N_NODES, E = N_EDGES;

    // workspace layout
    int*   deg_out   = (int*)d_ws;
    int*   deg_in    = deg_out + N;
    float* dnorm_out = (float*)d_ws + 2 * (size_t)N;
    float* dnorm_in  = (float*)d_ws + 3 * (size_t)N;
    float* agg       = (float*)d_ws + 4 * (size_t)N;    // N x 256 max
    float* hbuf      = agg  + (size_t)N * 256;          // N x 256
    float* advb      = hbuf + (size_t)N * 256;          // N x 64
    float* sums      = advb + (size_t)N * 64;           // 512
    float* coef      = sums + 512;                      // 512

    auto cdiv = [](long a, long b) { return (int)((a + b - 1) / b); };

    // ---- degrees / norms ----
    hipMemsetAsync(deg_out, 0, 2 * (size_t)N * sizeof(int), stream);
    degree_kernel<<<cdiv(E, 256), 256, 0, stream>>>(src, dst, deg_out, deg_in, E);
    dnorm_kernel<<<cdiv(N, 256), 256, 0, stream>>>(deg_out, deg_in, dnorm_out, dnorm_in, N);

    // ---- layer 1: F=128 -> 256 ----
    hipMemsetAsync(agg, 0, (size_t)N * 128 * sizeof(float), stream);
    spmm_scatter<128><<<cdiv((long)E * 32, 256), 256, 0, stream>>>(feat, src, dst, dnorm_out, agg, E);
    rowscale_kernel<128><<<cdiv((long)N * 32, 256), 256, 0, stream>>>(agg, dnorm_in, N);
    wmma_gemm_bias<128, 256><<<cdiv((long)(N / 16) * 16 * 32, 128), 128, 0, stream>>>(agg, W1, b1, hbuf, N);
    hipMemsetAsync(sums, 0, 512 * sizeof(float), stream);
    bn_stats_kernel<256, 512><<<cdiv(N, 512), 256, 0, stream>>>(hbuf, sums, N);
    bn_coef_kernel<<<1, 256, 0, stream>>>(sums, g1, be1, coef, N, 256);
    bn_apply_relu<256><<<cdiv((long)N * 256, 256), 256, 0, stream>>>(hbuf, coef, N);

    // ---- layer 2: F=256 -> 256 ----
    hipMemsetAsync(agg, 0, (size_t)N * 256 * sizeof(float), stream);
    spmm_scatter<256><<<cdiv((long)E * 32, 256), 256, 0, stream>>>(hbuf, src, dst, dnorm_out, agg, E);
    rowscale_kernel<256><<<cdiv((long)N * 64, 256), 256, 0, stream>>>(agg, dnorm_in, N);
    wmma_gemm_bias<256, 256><<<cdiv((long)(N / 16) * 16 * 32, 128), 128, 0, stream>>>(agg, W2, b2, hbuf, N);
    hipMemsetAsync(sums, 0, 512 * sizeof(float), stream);
    bn_stats_kernel<256, 512><<<cdiv(N, 512), 256, 0, stream>>>(hbuf, sums, N);
    bn_coef_kernel<<<1, 256, 0, stream>>>(sums, g2, be2, coef, N, 256);
    bn_apply_relu<256><<<cdiv((long)N * 256, 256), 256, 0, stream>>>(hbuf, coef, N);

    // ---- head ----
    wmma_gemm_bias<256, 64><<<cdiv((long)(N / 16) * 4 * 32, 128), 128, 0, stream>>>(hbuf, Wa, ba, advb, N);
    head_kernel<<<cdiv((long)N * 32, 256), 256, 0, stream>>>(hbuf, advb, Wv, bv, Q, N);
}